// GenerativeFusionCore_24343874634403
// MI455X (gfx1250) — compile-verified
//
#include <hip/hip_runtime.h>
#include <math.h>

#define NCAND 50000
#define EMB   512
#define HID   256
#define HID2  128
#define NGEN  20
#define CTXF  128

// padded LDS row strides (LDA/LDH/LDC == 4 mod 64 banks -> conflict-free A-fragment reads)
#define LDA   516   // candidate tile rows (512 cols)
#define LDH   260   // h rows (256 cols); also reused for f
#define LDC   132   // h2 rows / g rows (128 cols)
#define LDS_  36    // sims rows (32 cols)

typedef __attribute__((ext_vector_type(2))) float v2f;
typedef __attribute__((ext_vector_type(8))) float v8f;

__device__ __forceinline__ float sigmoidf_(float x) { return 1.0f / (1.0f + expf(-x)); }

__device__ __forceinline__ v8f wmma_step(v8f acc, float ax, float ay, float bx, float by) {
    v2f a; a.x = ax; a.y = ay;
    v2f b; b.x = bx; b.y = by;
    return __builtin_amdgcn_wmma_f32_16x16x4_f32(false, a, false, b, (short)0, acc, false, false);
}

// ---------------------------------------------------------------------------
// Kernel 0: meta-learner + precomputed constants (1 block, 256 threads)
// ws layout: [0..255] q_proj (incl nv_b1), [256..275] gen norms, [276] unorm
// ---------------------------------------------------------------------------
__global__ __launch_bounds__(256)
void gf_setup(const float* __restrict__ q, const float* __restrict__ user,
              const float* __restrict__ ctx, const float* __restrict__ trad,
              const float* __restrict__ gen,
              const float* __restrict__ mlw1, const float* __restrict__ mlb1,
              const float* __restrict__ mlw2, const float* __restrict__ mlb2,
              const float* __restrict__ nvw1, const float* __restrict__ nvb1,
              float* __restrict__ out, float* __restrict__ ws) {
    __shared__ float red[3 * 256];
    __shared__ float mean3[3];
    __shared__ float mlh[HID];
    __shared__ float lgs[3];
    const int t = threadIdx.x;

    float p0 = 0.f, p1 = 0.f, p2 = 0.f;
    for (int i = t; i < NCAND; i += 256) {
        p0 += trad[i * 3 + 0]; p1 += trad[i * 3 + 1]; p2 += trad[i * 3 + 2];
    }
    red[t] = p0; red[256 + t] = p1; red[512 + t] = p2;
    __syncthreads();
    for (int s = 128; s > 0; s >>= 1) {
        if (t < s) {
            red[t] += red[t + s];
            red[256 + t] += red[256 + t + s];
            red[512 + t] += red[512 + t + s];
        }
        __syncthreads();
    }
    if (t == 0) {
        mean3[0] = red[0]   * (1.0f / NCAND);
        mean3[1] = red[256] * (1.0f / NCAND);
        mean3[2] = red[512] * (1.0f / NCAND);
    }
    __syncthreads();

    {
        const int j = t;
        float h = mlb1[j];
        for (int i = 0; i < EMB; ++i) h += user[i] * mlw1[i * HID + j];
        h += mean3[0] * mlw1[(EMB + 0) * HID + j];
        h += mean3[1] * mlw1[(EMB + 1) * HID + j];
        h += mean3[2] * mlw1[(EMB + 2) * HID + j];
        for (int i = 0; i < CTXF; ++i) h += ctx[i] * mlw1[(EMB + 3 + i) * HID + j];
        mlh[j] = fmaxf(h, 0.f);
    }
    __syncthreads();
    if (t < 3) {
        float l = mlb2[t];
        for (int i = 0; i < HID; ++i) l += mlh[i] * mlw2[i * 3 + t];
        lgs[t] = l;
    }
    __syncthreads();
    if (t == 0) {
        float m = fmaxf(lgs[0], fmaxf(lgs[1], lgs[2]));
        float e0 = expf(lgs[0] - m), e1 = expf(lgs[1] - m), e2 = expf(lgs[2] - m);
        float d = e0 + e1 + e2;
        out[6 * NCAND + 0] = e0 / d;
        out[6 * NCAND + 1] = e1 / d;
        out[6 * NCAND + 2] = e2 / d;
    }

    {
        const int j = t;
        float v = nvb1[j];
        for (int k = 0; k < EMB; ++k) v += q[k] * nvw1[k * HID + j];
        ws[j] = v;
    }
    if (t < NGEN) {
        float s = 0.f;
        for (int k = 0; k < EMB; ++k) { float g = gen[t * EMB + k]; s += g * g; }
        ws[256 + t] = sqrtf(s);
    }
    if (t == 31) {
        float s = 0.f;
        for (int k = 0; k < EMB; ++k) { float u = user[k]; s += u * u; }
        ws[256 + NGEN] = sqrtf(s);
    }
}

// ---------------------------------------------------------------------------
// Main kernel: tile of 16 candidates per block, 256 threads (8 waves)
// ---------------------------------------------------------------------------
__global__ __launch_bounds__(256)
void gf_main(const float* __restrict__ cand, const float* __restrict__ user,
             const float* __restrict__ trad, const float* __restrict__ gen,
             const float* __restrict__ sfw1, const float* __restrict__ sfb1,
             const float* __restrict__ sfg1, const float* __restrict__ sfbe1,
             const float* __restrict__ sfw2, const float* __restrict__ sfb2,
             const float* __restrict__ sfg2, const float* __restrict__ sfbe2,
             const float* __restrict__ sfw3, const float* __restrict__ sfb3,
             const float* __restrict__ ccw1, const float* __restrict__ ccb1,
             const float* __restrict__ ccw2, const float* __restrict__ ccb2,
             const float* __restrict__ nvw1, const float* __restrict__ nvw2,
             const float* __restrict__ nvb2, const float* __restrict__ nvw3,
             const float* __restrict__ nvb3,
             float* __restrict__ out, const float* __restrict__ ws) {
    __shared__ float sA[16 * LDA];   // candidate tile; reused as f (16 x 256, stride LDH)
    __shared__ float sH[16 * LDH];   // novelty h; reused as g (16 x 128, stride LDC)
    __shared__ float sC[16 * LDC];   // novelty h2
    __shared__ float sS[16 * LDS_];  // raw sims dots
    __shared__ float qc[HID];
    __shared__ float gn[NGEN];
    __shared__ float cn[16], prs[16], s5[16 * 5];
    __shared__ float muS[16], rsS[16];
    __shared__ float un;

    const int t = threadIdx.x;
    const int wave = t >> 5;
    const int lane = t & 31;
    const int mn = lane & 15;          // M for A, N for B
    const int kh2 = (lane >> 4) << 1;  // K sub-offset: 0 or 2
    const int hh = lane >> 4;
    const int cand0 = blockIdx.x * 16;

    float* out_final = out;
    float* out_conf  = out + 1 * NCAND;
    float* out_nov   = out + 2 * NCAND;
    float* out_gen   = out + 3 * NCAND;
    float* out_gcf   = out + 4 * NCAND;
    float* out_pers  = out + 5 * NCAND;

    // ---- Phase 1: stage candidate tile (padded rows) + constants ----
    {
        const float4* src4 = (const float4*)(cand + (size_t)cand0 * EMB);
        for (int i = t; i < 16 * EMB / 4; i += 256) {
            int r = i >> 7, c = i & 127;  // 128 float4 per row
            *((float4*)(sA + r * LDA) + c) = src4[i];
        }
        qc[t] = ws[t];
        if (t < NGEN) gn[t] = ws[256 + t];
        if (t == 0) un = ws[256 + NGEN];
    }
    __syncthreads();

    // ---- Phase 2: sims WMMA (waves 0-1), GEMM1 WMMA (all), cnorm/pers (wave 7) ----
    if (t >= 224 && t < 240) {
        int r = t - 224;
        float ss = 0.f, du = 0.f;
        for (int k = 0; k < EMB; ++k) {
            float c = sA[r * LDA + k];
            ss += c * c;
            du += c * user[k];
        }
        float c2 = sqrtf(ss);
        cn[r] = c2;
        float p = du / fmaxf(c2 * un, 1e-8f);
        prs[r] = p;
        out_pers[cand0 + r] = p;
    }
    if (wave < 2) {
        // sims: clamp gen row index (cols >= NGEN produce garbage, never read)
        int n0 = wave * 16;
        int g = n0 + mn;
        const float* grow = gen + (size_t)(g < NGEN ? g : NGEN - 1) * EMB;
        const float* arow = sA + mn * LDA;
        v8f acc = {0.f, 0.f, 0.f, 0.f, 0.f, 0.f, 0.f, 0.f};
        #pragma unroll 4
        for (int k0 = 0; k0 < EMB; k0 += 4) {
            int k = k0 + kh2;
            float2 bv = *(const float2*)(grow + k);
            acc = wmma_step(acc, arow[k], arow[k + 1], bv.x, bv.y);
        }
        #pragma unroll
        for (int i = 0; i < 8; ++i) sS[(i + 8 * hh) * LDS_ + n0 + mn] = acc[i];
    }
    {   // GEMM1: h = relu(cand @ nv_w1[512:] + q_proj); 2 N-tiles per wave
        const float* B1 = nvw1 + EMB * HID;
        const float* arow = sA + mn * LDA;
        for (int tt = 0; tt < 2; ++tt) {
            int n0 = (wave * 2 + tt) * 16;
            int col = n0 + mn;
            v8f acc = {0.f, 0.f, 0.f, 0.f, 0.f, 0.f, 0.f, 0.f};
            #pragma unroll 4
            for (int k0 = 0; k0 < EMB; k0 += 4) {
                int k = k0 + kh2;
                acc = wmma_step(acc, arow[k], arow[k + 1],
                                B1[k * HID + col], B1[(k + 1) * HID + col]);
            }
            float qv = qc[col];
            #pragma unroll
            for (int i = 0; i < 8; ++i)
                sH[(i + 8 * hh) * LDH + col] = fmaxf(acc[i] + qv, 0.f);
        }
    }
    __syncthreads();

    // ---- Phase 3: generative max + unbiased std; assemble all_scores[16,5] ----
    if (t < 16) {
        int r = t;
        float sv[NGEN];
        float mx = -1e30f, sum = 0.f;
        float cr = cn[r];
        #pragma unroll
        for (int g = 0; g < NGEN; ++g) {
            float s = sS[r * LDS_ + g] / fmaxf(cr * gn[g], 1e-8f);
            sv[g] = s;
            mx = fmaxf(mx, s);
            sum += s;
        }
        float mu = sum * (1.0f / NGEN);
        float ss = 0.f;
        #pragma unroll
        for (int g = 0; g < NGEN; ++g) { float d = sv[g] - mu; ss += d * d; }
        out_gen[cand0 + r] = mx;
        out_gcf[cand0 + r] = sqrtf(ss * (1.0f / (NGEN - 1)));
        s5[r * 5 + 0] = trad[(size_t)(cand0 + r) * 3 + 0];
        s5[r * 5 + 1] = trad[(size_t)(cand0 + r) * 3 + 1];
        s5[r * 5 + 2] = trad[(size_t)(cand0 + r) * 3 + 2];
        s5[r * 5 + 3] = mx;
        s5[r * 5 + 4] = prs[r];
    }
    __syncthreads();

    // ---- Phase 4: sf layer1 (K=5, VALU) into sA; cc MLP; GEMM2 WMMA ----
    for (int idx = t; idx < 16 * HID; idx += 256) {
        int r = idx >> 8, j = idx & 255;
        const float* s = &s5[r * 5];
        float v = sfb1[j];
        #pragma unroll
        for (int k = 0; k < 5; ++k) v += s[k] * sfw1[k * HID + j];
        sA[r * LDH + j] = v;
    }
    if (t >= 16 && t < 32) {
        int r = t - 16;
        const float* s = &s5[r * 5];
        float a = 0.f;
        for (int j = 0; j < HID2; ++j) {
            float v = ccb1[j];
            #pragma unroll
            for (int k = 0; k < 5; ++k) v += s[k] * ccw1[k * HID2 + j];
            a += fmaxf(v, 0.f) * ccw2[j];
        }
        out_conf[cand0 + r] = sigmoidf_(a + ccb2[0]);
    }
    {   // GEMM2: h2 = relu(h @ nv_w2 + nv_b2); one 16-col tile per wave
        int n0 = wave * 16;
        int col = n0 + mn;
        const float* arow = sH + mn * LDH;
        v8f acc = {0.f, 0.f, 0.f, 0.f, 0.f, 0.f, 0.f, 0.f};
        #pragma unroll 4
        for (int k0 = 0; k0 < HID; k0 += 4) {
            int k = k0 + kh2;
            acc = wmma_step(acc, arow[k], arow[k + 1],
                            nvw2[k * HID2 + col], nvw2[(k + 1) * HID2 + col]);
        }
        float bv = nvb2[col];
        #pragma unroll
        for (int i = 0; i < 8; ++i)
            sC[(i + 8 * hh) * LDC + col] = fmaxf(acc[i] + bv, 0.f);
    }
    __syncthreads();

    // ---- Phase 5: LN1 stats; novelty output; apply LN1+relu ----
    if (t < 16) {
        int r = t;
        float sum = 0.f;
        for (int j = 0; j < HID; ++j) sum += sA[r * LDH + j];
        float mu = sum * (1.0f / HID);
        float vs = 0.f;
        for (int j = 0; j < HID; ++j) { float d = sA[r * LDH + j] - mu; vs += d * d; }
        muS[r] = mu;
        rsS[r] = rsqrtf(vs * (1.0f / HID) + 1e-5f);
    }
    if (t >= 32 && t < 48) {
        int r = t - 32;
        float a = nvb3[0];
        for (int j = 0; j < HID2; ++j) a += sC[r * LDC + j] * nvw3[j];
        out_nov[cand0 + r] = sigmoidf_(a);
    }
    __syncthreads();
    for (int idx = t; idx < 16 * HID; idx += 256) {
        int r = idx >> 8, j = idx & 255;
        float x = (sA[r * LDH + j] - muS[r]) * rsS[r] * sfg1[j] + sfbe1[j];
        sA[r * LDH + j] = fmaxf(x, 0.f);
    }
    __syncthreads();

    // ---- Phase 6: sf layer2 WMMA: g = f @ sf_w2 + b2 into sH (stride LDC) ----
    {
        int n0 = wave * 16;
        int col = n0 + mn;
        const float* arow = sA + mn * LDH;
        v8f acc = {0.f, 0.f, 0.f, 0.f, 0.f, 0.f, 0.f, 0.f};
        #pragma unroll 4
        for (int k0 = 0; k0 < HID; k0 += 4) {
            int k = k0 + kh2;
            acc = wmma_step(acc, arow[k], arow[k + 1],
                            sfw2[k * HID2 + col], sfw2[(k + 1) * HID2 + col]);
        }
        float bv = sfb2[col];
        #pragma unroll
        for (int i = 0; i < 8; ++i)
            sH[(i + 8 * hh) * LDC + col] = acc[i] + bv;
    }
    __syncthreads();

    // ---- Phase 7: LN2 + relu + final head ----
    if (t < 16) {
        int r = t;
        float sum = 0.f;
        for (int j = 0; j < HID2; ++j) sum += sH[r * LDC + j];
        float mu = sum * (1.0f / HID2);
        float vs = 0.f;
        for (int j = 0; j < HID2; ++j) { float d = sH[r * LDC + j] - mu; vs += d * d; }
        muS[r] = mu;
        rsS[r] = rsqrtf(vs * (1.0f / HID2) + 1e-5f);
    }
    __syncthreads();
    for (int idx = t; idx < 16 * HID2; idx += 256) {
        int r = idx >> 7, j = idx & 127;
        float x = (sH[r * LDC + j] - muS[r]) * rsS[r] * sfg2[j] + sfbe2[j];
        sH[r * LDC + j] = fmaxf(x, 0.f);
    }
    __syncthreads();
    if (t < 16) {
        int r = t;
        float a = sfb3[0];
        for (int j = 0; j < HID2; ++j) a += sH[r * LDC + j] * sfw3[j];
        out_final[cand0 + r] = sigmoidf_(a);
    }
}

extern "C" void kernel_launch(void* const* d_in, const int* in_sizes, int n_in,
                              void* d_out, int out_size, void* d_ws, size_t ws_size,
                              hipStream_t stream) {
    const float* q    = (const float*)d_in[0];
    const float* cand = (const float*)d_in[1];
    const float* user = (const float*)d_in[2];
    const float* ctx  = (const float*)d_in[3];
    const float* trad = (const float*)d_in[4];
    const float* gen  = (const float*)d_in[5];
    const float* mlw1 = (const float*)d_in[6];
    const float* mlb1 = (const float*)d_in[7];
    const float* mlw2 = (const float*)d_in[8];
    const float* mlb2 = (const float*)d_in[9];
    const float* sfw1 = (const float*)d_in[10];
    const float* sfb1 = (const float*)d_in[11];
    const float* sfg1 = (const float*)d_in[12];
    const float* sfbe1 = (const float*)d_in[13];
    const float* sfw2 = (const float*)d_in[14];
    const float* sfb2 = (const float*)d_in[15];
    const float* sfg2 = (const float*)d_in[16];
    const float* sfbe2 = (const float*)d_in[17];
    const float* sfw3 = (const float*)d_in[18];
    const float* sfb3 = (const float*)d_in[19];
    const float* ccw1 = (const float*)d_in[20];
    const float* ccb1 = (const float*)d_in[21];
    const float* ccw2 = (const float*)d_in[22];
    const float* ccb2 = (const float*)d_in[23];
    const float* nvw1 = (const float*)d_in[24];
    const float* nvb1 = (const float*)d_in[25];
    const float* nvw2 = (const float*)d_in[26];
    const float* nvb2 = (const float*)d_in[27];
    const float* nvw3 = (const float*)d_in[28];
    const float* nvb3 = (const float*)d_in[29];
    float* out = (float*)d_out;
    float* ws  = (float*)d_ws;

    gf_setup<<<1, 256, 0, stream>>>(q, user, ctx, trad, gen,
                                    mlw1, mlb1, mlw2, mlb2, nvw1, nvb1, out, ws);
    gf_main<<<NCAND / 16, 256, 0, stream>>>(cand, user, trad, gen,
                                            sfw1, sfb1, sfg1, sfbe1,
                                            sfw2, sfb2, sfg2, sfbe2,
                                            sfw3, sfb3,
                                            ccw1, ccb1, ccw2, ccb2,
                                            nvw1, nvw2, nvb2, nvw3, nvb3,
                                            out, ws);
}